// Universal_48438641164344
// MI455X (gfx1250) — compile-verified
//
#include <hip/hip_runtime.h>

// ---------------------------------------------------------------------------
// GCN pipeline for MI455X (gfx1250, wave32, WMMA).
//   h   = relu(x@W1+b1)@W2+b2                         [N,16]
//   h   = diffuse(h)  (20 steps, edge scatter+update)
//   out = per-class heads fused GEMM (x@A1w[c,1:]) + relu + dot(A2w)
//   out = diffuse(out)
// fp32 WMMA: V_WMMA_F32_16X16X4_F32, D = A(16x4) x B(4x16) + C(16x16).
// ---------------------------------------------------------------------------

#define NN 100000
#define NE 3200000
#define FEATS 256
#define CLS 16
#define HID 64
#define NDEPTH 20
#define DIFFC 0.9f
#define NTILES (NN / 16)

typedef __attribute__((ext_vector_type(2))) float v2f;
typedef __attribute__((ext_vector_type(8))) float v8f;

__device__ __forceinline__ v8f wmma_f32(v2f a, v2f b, v8f c) {
    // 8 args: (neg_a, A, neg_b, B, c_mod, C, reuse_a, reuse_b)
    return __builtin_amdgcn_wmma_f32_16x16x4_f32(false, a, false, b, (short)0, c,
                                                 false, false);
}

// --------------------------- graph norm kernels ----------------------------

__global__ void k_init(float* __restrict__ deg, float* __restrict__ acc) {
    int i = blockIdx.x * blockDim.x + threadIdx.x;
    if (i < NN * CLS) acc[i] = 0.0f;
    if (i < NN) deg[i] = 0.0f;
}

__global__ void k_deg(const int* __restrict__ edges, float* __restrict__ deg) {
    int e = blockIdx.x * blockDim.x + threadIdx.x;
    if (e >= NE) return;
    atomicAdd(&deg[edges[NE + e]], 1.0f);
}

__global__ void k_dis(float* __restrict__ deg) {
    int i = blockIdx.x * blockDim.x + threadIdx.x;
    if (i >= NN) return;
    float d = deg[i];
    deg[i] = (d > 0.0f) ? rsqrtf(fmaxf(d, 1.0f)) : 0.0f;
}

__global__ void k_norm(const int* __restrict__ edges, const float* __restrict__ dis,
                       float* __restrict__ nrm) {
    int e = blockIdx.x * blockDim.x + threadIdx.x;
    if (e >= NE) return;
    nrm[e] = dis[edges[e]] * dis[edges[NE + e]];
}

// --------------------------- diffusion kernels -----------------------------

__global__ void k_scatter(const float* __restrict__ h, const int* __restrict__ edges,
                          const float* __restrict__ nrm, float* __restrict__ acc) {
    int e = blockIdx.x * blockDim.x + threadIdx.x;
    if (e >= NE) return;
    int s = edges[e];
    int d = edges[NE + e];
    float w = nrm[e];
    const float4* hs = (const float4*)(h + (size_t)s * CLS);
    float* ad = acc + (size_t)d * CLS;
#pragma unroll
    for (int j = 0; j < 4; ++j) {
        float4 v = hs[j];
        atomicAdd(ad + 4 * j + 0, v.x * w);
        atomicAdd(ad + 4 * j + 1, v.y * w);
        atomicAdd(ad + 4 * j + 2, v.z * w);
        atomicAdd(ad + 4 * j + 3, v.w * w);
    }
}

__global__ void k_update(float* __restrict__ cur, const float* __restrict__ h0,
                         float* __restrict__ acc) {
    int i = blockIdx.x * blockDim.x + threadIdx.x;
    if (i >= NN * CLS) return;
    cur[i] = acc[i] * DIFFC + (1.0f - DIFFC) * h0[i];
    acc[i] = 0.0f;  // ready for next scatter
}

// --------------------------- MLP: h = relu(x@W1+b1)@W2+b2 ------------------
// One wave (32 lanes) per 16-node tile. 4 waves / block.

__global__ void __launch_bounds__(128) k_mlp(const float* __restrict__ x,
                                             const float* __restrict__ W1,
                                             const float* __restrict__ b1,
                                             const float* __restrict__ W2,
                                             const float* __restrict__ b2,
                                             float* __restrict__ h0,
                                             float* __restrict__ cur) {
    __shared__ float lds[4 * 16 * HID];
    int lane = threadIdx.x & 31;
    int wv = threadIdx.x >> 5;
    int tile = blockIdx.x * 4 + wv;
    if (tile >= NTILES) return;  // wave-uniform: EXEC stays all-1s for WMMA
    int row0 = tile * 16;
    int half = lane >> 4;
    int l16 = lane & 15;

    v8f zero = {0.f, 0.f, 0.f, 0.f, 0.f, 0.f, 0.f, 0.f};
    v8f acc[4] = {zero, zero, zero, zero};

    // GEMM1: x[16,256] @ W1[256,64] -> 16x64, K-step = 4
    const float* xrow = x + (size_t)(row0 + l16) * FEATS + 2 * half;
    for (int k = 0; k < FEATS; k += 4) {
        v2f a;
        a.x = xrow[k];
        a.y = xrow[k + 1];
#pragma unroll
        for (int t = 0; t < 4; ++t) {
            v2f b;
            b.x = W1[(k + half) * HID + t * 16 + l16];
            b.y = W1[(k + 2 + half) * HID + t * 16 + l16];
            acc[t] = wmma_f32(a, b, acc[t]);
        }
    }

    // bias + relu, stage tile into LDS (C layout -> row major)
    float* myl = lds + wv * 16 * HID;
#pragma unroll
    for (int t = 0; t < 4; ++t) {
#pragma unroll
        for (int i = 0; i < 8; ++i) {
            int r = i + 8 * half;
            int hc = t * 16 + l16;
            myl[r * HID + hc] = fmaxf(acc[t][i] + b1[hc], 0.0f);
        }
    }
    asm volatile("s_wait_dscnt 0" ::: "memory");  // wave-local LDS RAW fence

    // GEMM2: h1[16,64] @ W2[64,16] -> 16x16
    v8f acc2 = zero;
    for (int k = 0; k < HID; k += 4) {
        v2f a;
        a.x = myl[l16 * HID + k + 2 * half];
        a.y = myl[l16 * HID + k + 2 * half + 1];
        v2f b;
        b.x = W2[(k + half) * CLS + l16];
        b.y = W2[(k + 2 + half) * CLS + l16];
        acc2 = wmma_f32(a, b, acc2);
    }
#pragma unroll
    for (int i = 0; i < 8; ++i) {
        int r = row0 + i + 8 * half;
        float v = acc2[i] + b2[l16];
        h0[(size_t)r * CLS + l16] = v;
        cur[(size_t)r * CLS + l16] = v;
    }
}

// ----------- fused per-class heads: out[n,c] = relu(x@A1w'+...)·A2w --------
// x tile A-fragments cached in registers (64 x v2f) and reused for all 16
// classes, so x is streamed from HBM exactly once.

__global__ void __launch_bounds__(128) k_heads(const float* __restrict__ x,
                                               const float* __restrict__ hdiff,
                                               const float* __restrict__ A1w,
                                               const float* __restrict__ A1b,
                                               const float* __restrict__ A2w,
                                               const float* __restrict__ A2b,
                                               float* __restrict__ out0,
                                               float* __restrict__ cur2) {
    int lane = threadIdx.x & 31;
    int wv = threadIdx.x >> 5;
    int tile = blockIdx.x * 4 + wv;
    if (tile >= NTILES) return;
    int row0 = tile * 16;
    int half = lane >> 4;
    int l16 = lane & 15;

    const float* xrow = x + (size_t)(row0 + l16) * FEATS + 2 * half;
    v2f af[FEATS / 4];
#pragma unroll
    for (int ks = 0; ks < FEATS / 4; ++ks) {
        af[ks].x = xrow[4 * ks];
        af[ks].y = xrow[4 * ks + 1];
    }

    v8f zero = {0.f, 0.f, 0.f, 0.f, 0.f, 0.f, 0.f, 0.f};

#pragma unroll 1
    for (int c = 0; c < CLS; ++c) {
        const float* Wc = A1w + (size_t)c * (1 + FEATS) * HID;  // f=0 row: h-channel weight
        v8f acc[4] = {zero, zero, zero, zero};
#pragma unroll
        for (int ks = 0; ks < FEATS / 4; ++ks) {
            int k = 4 * ks;
#pragma unroll
            for (int t = 0; t < 4; ++t) {
                v2f b;
                b.x = Wc[(size_t)(1 + k + half) * HID + t * 16 + l16];
                b.y = Wc[(size_t)(3 + k + half) * HID + t * 16 + l16];
                acc[t] = wmma_f32(af[ks], b, acc[t]);
            }
        }
        // epilogue: z = relu(acc + hdiff[n,c]*A1w[c,0,:] + A1b[c]); out = z·A2w[c]
        float hv[8];
#pragma unroll
        for (int i = 0; i < 8; ++i)
            hv[i] = hdiff[(size_t)(row0 + i + 8 * half) * CLS + c];
        float part[8];
#pragma unroll
        for (int i = 0; i < 8; ++i) part[i] = 0.0f;
#pragma unroll
        for (int t = 0; t < 4; ++t) {
            int hc = t * 16 + l16;
            float w0 = Wc[hc];
            float bb = A1b[c * HID + hc];
            float a2 = A2w[c * HID + hc];
#pragma unroll
            for (int i = 0; i < 8; ++i) {
                float z = fmaxf(acc[t][i] + hv[i] * w0 + bb, 0.0f);
                part[i] += z * a2;
            }
        }
        // reduce the 64 hidden dims: 4 tiles done above, now 16 lanes per half
#pragma unroll
        for (int off = 1; off < 16; off <<= 1) {
#pragma unroll
            for (int i = 0; i < 8; ++i) part[i] += __shfl_xor(part[i], off, 32);
        }
        if (l16 == 0) {
#pragma unroll
            for (int i = 0; i < 8; ++i) {
                float v = part[i] + A2b[c];
                size_t idx = (size_t)(row0 + i + 8 * half) * CLS + c;
                out0[idx] = v;
                cur2[idx] = v;
            }
        }
    }
}

// ---------------------------------------------------------------------------

extern "C" void kernel_launch(void* const* d_in, const int* in_sizes, int n_in,
                              void* d_out, int out_size, void* d_ws, size_t ws_size,
                              hipStream_t stream) {
    const float* x   = (const float*)d_in[0];
    const int* edges = (const int*)d_in[1];
    const float* W1  = (const float*)d_in[2];
    const float* b1  = (const float*)d_in[3];
    const float* W2  = (const float*)d_in[4];
    const float* b2  = (const float*)d_in[5];
    const float* A1w = (const float*)d_in[6];
    const float* A1b = (const float*)d_in[7];
    const float* A2w = (const float*)d_in[8];
    const float* A2b = (const float*)d_in[9];

    float* ws = (float*)d_ws;
    float* deg  = ws;                               // N  (becomes dis)
    float* nrm  = deg + NN;                         // E
    float* h0   = nrm + NE;                         // N*C
    float* cur  = h0 + (size_t)NN * CLS;            // N*C
    float* acc  = cur + (size_t)NN * CLS;           // N*C
    float* out0 = acc + (size_t)NN * CLS;           // N*C

    const int TB = 256;
    int gNC = (NN * CLS + TB - 1) / TB;
    int gE  = (NE + TB - 1) / TB;
    int gN  = (NN + TB - 1) / TB;
    int gT  = (NTILES + 3) / 4;

    k_init<<<gNC, TB, 0, stream>>>(deg, acc);
    k_deg<<<gE, TB, 0, stream>>>(edges, deg);
    k_dis<<<gN, TB, 0, stream>>>(deg);
    k_norm<<<gE, TB, 0, stream>>>(edges, deg, nrm);

    k_mlp<<<gT, 128, 0, stream>>>(x, W1, b1, W2, b2, h0, cur);

    for (int it = 0; it < NDEPTH; ++it) {
        k_scatter<<<gE, TB, 0, stream>>>(cur, edges, nrm, acc);
        k_update<<<gNC, TB, 0, stream>>>(cur, h0, acc);
    }

    // heads: writes out0 (baseline for 2nd diffusion) and cur2 (state, reuses h0 buf)
    k_heads<<<gT, 128, 0, stream>>>(x, cur, A1w, A1b, A2w, A2b, out0, h0);

    for (int it = 0; it < NDEPTH; ++it) {
        k_scatter<<<gE, TB, 0, stream>>>(h0, edges, nrm, acc);
        k_update<<<gNC, TB, 0, stream>>>(h0, out0, acc);
    }

    hipMemcpyAsync(d_out, h0, (size_t)NN * CLS * sizeof(float),
                   hipMemcpyDeviceToDevice, stream);
}